// Model_83047487636191
// MI455X (gfx1250) — compile-verified
//
#include <hip/hip_runtime.h>
#include <hip/hip_bf16.h>

// ---------------- problem constants ----------------
#define NUMPTS 53149
#define NROWS  (NUMPTS * 3)      // 159447
#define KPCA   80
#define BATCH  16
#define HIMG   640
#define WIMG   360
#define PIXB   (HIMG * WIMG)     // 230400
#define NPIX   (BATCH * PIXB)    // 3686400
#define GATHER_IDX 26574         // constant iy of the degenerate grid_sample

// ---------------- output layout (floats) ----------------
#define O_LK    0
#define O_LK1   1
#define O_LREG  2
#define O_LD    3
#define O_R1    4                // 144
#define O_T     148              // 48
#define O_PK2   196              // 2176
#define O_MASK  2372             // 3686400 (also reused as scatter buffer)
#define O_PRED  (O_MASK + NPIX)  // 3686400

// ---------------- small workspace layout (floats) ----------------
#define WS_COEFF 0               // 80
#define WS_R1    80              // 144
#define WS_PRED  224             // 16
#define WS_LDSUM 240             // 16
#define WS_CNT   256             // 16

typedef float v2f __attribute__((ext_vector_type(2)));
typedef float v8f __attribute__((ext_vector_type(8)));

// Order-preserving float<->int map (total order matches IEEE for non-NaN),
// so scatter-max can use native global_atomic_max_i32.
__device__ __forceinline__ int encf(float f) {
    int i = __float_as_int(f);
    return (i >= 0) ? i : (i ^ 0x7FFFFFFF);
}
__device__ __forceinline__ float decf(int i) {
    return __int_as_float((i >= 0) ? i : (i ^ 0x7FFFFFFF));
}

// ============================================================
// Kernel 0: initialize scatter buffer with enc(-10.0f)
// ============================================================
__global__ __launch_bounds__(256) void init_scat(int* __restrict__ scat) {
    int i = blockIdx.x * 256 + threadIdx.x;
    if (i < NPIX) scat[i] = encf(-10.0f);
}

// ============================================================
// Kernel 1 (tiny, 1 block): coeff, R1, T passthrough, keypoint
// losses, loss_reg, accumulator zeroing.
// ============================================================
__global__ __launch_bounds__(256) void setup_kernel(
    const float* __restrict__ mean, const float* __restrict__ pca,
    const float* __restrict__ variance, const float* __restrict__ param,
    const float* __restrict__ R, const float* __restrict__ T,
    const int* __restrict__ key_keypoint,
    const float* __restrict__ key_ref, const float* __restrict__ key_side_ref,
    float* __restrict__ out, float* __restrict__ wss)
{
    __shared__ float s_coeff[KPCA];
    __shared__ float s_R1[BATCH * 9];
    __shared__ float s_pk[68 * 3];
    __shared__ float s_pk2[BATCH * 68 * 2];

    const int tid = threadIdx.x;

    if (tid < KPCA) {
        float c = param[tid] * sqrtf(variance[tid]);
        s_coeff[tid] = c;
        wss[WS_COEFF + tid] = c;
    }
    if (tid < BATCH) { wss[WS_LDSUM + tid] = 0.0f; wss[WS_CNT + tid] = 0.0f; }
    __syncthreads();

    // Rz @ Ry @ Rx per batch
    if (tid < BATCH) {
        float ax = R[tid * 3 + 0], ay = R[tid * 3 + 1], az = R[tid * 3 + 2];
        float sx = sinf(ax), cx = cosf(ax);
        float sy = sinf(ay), cy = cosf(ay);
        float sz = sinf(az), cz = cosf(az);
        float m[9] = { cz * cy, -sz * cx + cz * sy * sx,  sz * sx + cz * sy * cx,
                       sz * cy,  cz * cx + sz * sy * sx, -cz * sx + sz * sy * cx,
                       -sy,      cy * sx,                 cy * cx };
        for (int i = 0; i < 9; ++i) {
            s_R1[tid * 9 + i] = m[i];
            out[O_R1 + tid * 9 + i] = m[i];
            wss[WS_R1 + tid * 9 + i] = m[i];
        }
    }
    if (tid < BATCH * 3) out[O_T + tid] = T[tid];

    // keypoint model points: 68 keypoints x 3 coords = 204 dot products of 80
    if (tid < 68 * 3) {
        int k = tid / 3, c = tid % 3;
        int row = key_keypoint[k] * 3 + c;
        const float* a = pca + (long)row * KPCA;
        float s = 0.0f;
        for (int j = 0; j < KPCA; ++j) s += a[j] * s_coeff[j];
        s_pk[tid] = s + mean[row];
    }
    __syncthreads();

    // rotate + project keypoints: pk2 (B,68,2)
    for (int i = tid; i < BATCH * 68; i += 256) {
        int b = i / 68, k = i % 68;
        float x = s_pk[k * 3 + 0], y = s_pk[k * 3 + 1], z = s_pk[k * 3 + 2];
        const float* Rb = &s_R1[b * 9];
        float q0 = x * Rb[0] + y * Rb[3] + z * Rb[6] + T[b * 3 + 0];
        float q1 = x * Rb[1] + y * Rb[4] + z * Rb[7] + T[b * 3 + 1];
        float q2 = x * Rb[2] + y * Rb[5] + z * Rb[8] + T[b * 3 + 2];
        float pz = -q2;
        float px = (436.16f * q0 + 179.22f * pz) / pz;
        float py = (436.16f * q1 + 320.08f * pz) / pz;
        s_pk2[i * 2 + 0] = px; s_pk2[i * 2 + 1] = py;
        out[O_PK2 + i * 2 + 0] = px; out[O_PK2 + i * 2 + 1] = py;
    }
    __syncthreads();

    if (tid == 0) {
        float sr = 0.0f;
        for (int j = 0; j < KPCA; ++j) { float p = param[j]; sr += p * p; }
        out[O_LREG] = sr / (float)KPCA;

        // INNER = [8, 17..67]
        float sk = 0.0f;
        for (int b = 0; b < BATCH; ++b)
            for (int i = 0; i < 52; ++i) {
                int k = (i == 0) ? 8 : (i + 16);
                float dx = s_pk2[(b * 68 + k) * 2 + 0] - key_ref[(b * 52 + i) * 2 + 0];
                float dy = s_pk2[(b * 68 + k) * 2 + 1] - key_ref[(b * 52 + i) * 2 + 1];
                sk += dx * dx + dy * dy;
            }
        out[O_LK] = sk / (float)(BATCH * 52 * 2);

        // SIDE = [0..7, 9..16]
        float s1 = 0.0f;
        for (int b = 0; b < BATCH; ++b)
            for (int i = 0; i < 16; ++i) {
                int k = (i < 8) ? i : (i + 1);
                float dx = s_pk2[(b * 68 + k) * 2 + 0] - key_side_ref[(b * 16 + i) * 2 + 0];
                float dy = s_pk2[(b * 68 + k) * 2 + 1] - key_side_ref[(b * 16 + i) * 2 + 1];
                s1 += dx * dx + dy * dy;
            }
        out[O_LK1] = s1 / (float)(BATCH * 16 * 2);
    }
}

// ============================================================
// Kernel 2: WMMA GEMV (pca @ coeff) fused with per-batch
// transform + projection + scatter-max.  Never materializes
// points0/pts/p1 in HBM.
//   block = 256 threads = 8 waves; block covers 256 points
//   = 768 flat rows = 48 row-tiles of 16; 6 tiles per wave.
//   Each tile: 20 x V_WMMA_F32_16X16X4_F32 over K=80, with the
//   coefficient vector replicated across all 16 B columns.
// ============================================================
#define ROWS_PER_BLOCK 768
#define TILES_PER_WAVE 6

__global__ __launch_bounds__(256) void pca_scatter_kernel(
    const float* __restrict__ pca, const float* __restrict__ mean,
    const float* __restrict__ T, const float* __restrict__ wss,
    int* __restrict__ scat, float* __restrict__ pred_scalar)
{
    __shared__ float sh_coeff[KPCA];
    __shared__ float sh_R1[BATCH * 9];
    __shared__ float sh_T[BATCH * 3];
    __shared__ float sh_pts[ROWS_PER_BLOCK];

    const int tid = threadIdx.x;
    if (tid < KPCA)                    sh_coeff[tid]      = wss[WS_COEFF + tid];
    if (tid >= 128 && tid < 128 + 144) sh_R1[tid - 128]   = wss[WS_R1 + tid - 128];
    if (tid >= 80 && tid < 80 + 48)    sh_T[tid - 80]     = T[tid - 80];
    __syncthreads();

    const int lane = tid & 31;
    const int wave = tid >> 5;
    const int half = lane >> 4;    // 0: K pair {0,1}, rows r;  1: K pair {2,3}, rows r+8
    const int lrow = lane & 15;

    const int baseRow = blockIdx.x * ROWS_PER_BLOCK;

    for (int t = 0; t < TILES_PER_WAVE; ++t) {
        const int tileRow = wave * (TILES_PER_WAVE * 16) + t * 16;  // local row of tile
        int gr = baseRow + tileRow + lrow;
        if (gr > NROWS - 1) gr = NROWS - 1;                          // clamp OOB lanes
        const float* arow = pca + (long)gr * KPCA;
        __builtin_prefetch(arow + 16 * KPCA, 0, 1);                  // next tile's row

        v8f acc = {};
#pragma unroll
        for (int s = 0; s < 20; ++s) {
            const int kk = 4 * s + 2 * half;
            v2f a;  a.x = arow[kk];       a.y = arow[kk + 1];
            v2f bb; bb.x = sh_coeff[kk];  bb.y = sh_coeff[kk + 1];   // all 16 cols identical
            acc = __builtin_amdgcn_wmma_f32_16x16x4_f32(
                false, a, false, bb, (short)0, acc, false, false);
        }
        // D layout: lane-half 0 holds rows r (c[r]); lane-half 1 holds rows r+8.
        if (lrow == 0) {
#pragma unroll
            for (int r = 0; r < 8; ++r) {
                const int lr = tileRow + half * 8 + r;
                const int grr = baseRow + lr;
                const float m = (grr < NROWS) ? mean[grr] : 0.0f;
                sh_pts[lr] = acc[r] + m;
            }
        }
    }
    __syncthreads();

    // transform + project + scatter-max (16 batches per point)
    const int n = blockIdx.x * 256 + tid;
    if (n < NUMPTS) {
        const float x = sh_pts[tid * 3 + 0];
        const float y = sh_pts[tid * 3 + 1];
        const float z = sh_pts[tid * 3 + 2];
        const float val = ((float)n / (float)NUMPTS) * 2.0f - 1.0f;
        const int enc_val = encf(val);
#pragma unroll 1
        for (int b = 0; b < BATCH; ++b) {
            const float* Rb = &sh_R1[b * 9];
            const float* Tb = &sh_T[b * 3];
            float q0 = x * Rb[0] + y * Rb[3] + z * Rb[6] + Tb[0];
            float q1 = x * Rb[1] + y * Rb[4] + z * Rb[7] + Tb[1];
            float q2 = x * Rb[2] + y * Rb[5] + z * Rb[8] + Tb[2];
            float pz = -q2;
            float p1x = (436.16f * q0 + 179.22f * pz) / pz;
            float p1y = (436.16f * q1 + 320.08f * pz) / pz;
            bool m0 = (p1y >= 0.0f) && (p1y <= (float)(HIMG - 1)) &&
                      (p1x >= 0.0f) && (p1x <= (float)(WIMG - 1));
            p1x = m0 ? p1x : 0.0f;
            p1y = m0 ? p1y : 0.0f;
            const int xi = (int)p1x;      // trunc toward zero, matches astype(int32)
            const int yi = (int)p1y;
            atomicMax(&scat[b * PIXB + yi * WIMG + xi], enc_val);
            if (n == GATHER_IDX) pred_scalar[b] = p1y;  // masked y-coord @ index 26574
        }
    }
}

// ============================================================
// Kernel 3: per-pixel depth loss.  scat (encoded) lives in the
// mask output region and is overwritten in place (1:1 mapping).
// ============================================================
__global__ __launch_bounds__(256) void depth_kernel(
    const float* __restrict__ ref_depth, const float* __restrict__ pred_scalar,
    const int* __restrict__ dparam,
    int* scat_and_mask,              // aliases out+O_MASK: read enc, write mask
    float* __restrict__ out_pred, float* __restrict__ wss)
{
    __shared__ float red[256];

    const int tid = threadIdx.x;
    const int b   = blockIdx.y;
    const int pix = blockIdx.x * 256 + tid;
    const int idx = b * PIXB + pix;
    const float dthr = (float)dparam[0];

    const float s   = decf(scat_and_mask[idx]);
    const float ch0 = (s > -5.0f) ? s : 1.0f;
    const bool  m_pix = (ch0 >= -1.0f) && (ch0 < 1.0f);
    const float mf = m_pix ? 1.0f : 0.0f;
    // grid_sample collapses: W=1, gy=0 -> iy==26574 const; valid iff ix==0
    const int   ix = (int)rintf((ch0 + 1.0f) * 0.5f - 0.5f);
    const float gs = pred_scalar[b] * ((ix == 0) ? 1.0f : 0.0f);
    const float pred = gs * mf;
    const float rd = ref_depth[idx];
    const float diff = pred - rd * mf;
    const float ld = diff * diff;
    const bool  msk = (ld < dthr) && (ld > 1e-6f) && (pred > 0.0f);
    const float ldo = ((ld > dthr) || (pred < 1e-5f)) ? 0.0f : ld;

    scat_and_mask[idx] = __float_as_int(msk ? 1.0f : 0.0f);  // mask output (float bits)
    out_pred[idx] = pred;

    // block reduction: sum(ldo), count(msk)
    red[tid] = ldo;
    __syncthreads();
    for (int o = 128; o > 0; o >>= 1) {
        if (tid < o) red[tid] += red[tid + o];
        __syncthreads();
    }
    if (tid == 0) atomicAdd(&wss[WS_LDSUM + b], red[0]);
    __syncthreads();
    red[tid] = msk ? 1.0f : 0.0f;
    __syncthreads();
    for (int o = 128; o > 0; o >>= 1) {
        if (tid < o) red[tid] += red[tid + o];
        __syncthreads();
    }
    if (tid == 0) atomicAdd(&wss[WS_CNT + b], red[0]);
}

// ============================================================
// Kernel 4: finalize loss_d
// ============================================================
__global__ void finalize_kernel(const float* __restrict__ wss, float* __restrict__ out) {
    if (threadIdx.x == 0) {
        float s = 0.0f;
        for (int b = 0; b < BATCH; ++b)
            s += wss[WS_LDSUM + b] / (wss[WS_CNT + b] + 1.0f);
        out[O_LD] = s / (float)BATCH;
    }
}

// ============================================================
extern "C" void kernel_launch(void* const* d_in, const int* in_sizes, int n_in,
                              void* d_out, int out_size, void* d_ws, size_t ws_size,
                              hipStream_t stream) {
    (void)in_sizes; (void)n_in; (void)out_size; (void)ws_size;
    const float* mean         = (const float*)d_in[0];
    const float* pca          = (const float*)d_in[1];
    const float* variance     = (const float*)d_in[2];
    const float* param        = (const float*)d_in[3];
    const float* R            = (const float*)d_in[4];
    const float* T            = (const float*)d_in[5];
    const int*   key_keypoint = (const int*)d_in[6];
    const float* key_ref      = (const float*)d_in[7];
    const float* key_side_ref = (const float*)d_in[8];
    const float* ref_depth    = (const float*)d_in[9];
    const int*   dparam       = (const int*)d_in[10];

    float* out = (float*)d_out;
    float* wss = (float*)d_ws;
    int*   scat = (int*)(out + O_MASK);   // reuse mask region as scatter buffer

    init_scat<<<(NPIX + 255) / 256, 256, 0, stream>>>(scat);
    setup_kernel<<<1, 256, 0, stream>>>(mean, pca, variance, param, R, T,
                                        key_keypoint, key_ref, key_side_ref, out, wss);
    const int nblk = (NROWS + ROWS_PER_BLOCK - 1) / ROWS_PER_BLOCK;  // 208
    pca_scatter_kernel<<<nblk, 256, 0, stream>>>(pca, mean, T, wss, scat, wss + WS_PRED);
    depth_kernel<<<dim3(PIXB / 256, BATCH), 256, 0, stream>>>(
        ref_depth, wss + WS_PRED, dparam, scat, out + O_PRED, wss);
    finalize_kernel<<<1, 32, 0, stream>>>(wss, out);
}